// CircuitRouter_55095840473243
// MI455X (gfx1250) — compile-verified
//
#include <hip/hip_runtime.h>
#include <math.h>

// Problem constants (fixed by reference setup_inputs)
#define B_ROWS  4096
#define D_MODEL 1024
#define D_SPACE 64
#define NC_R    64
#define N_NEUR  64
#define TOPK    8
#define TAU_INV (1.0f / 1.5f)

typedef float v2f __attribute__((ext_vector_type(2)));
typedef float v8f __attribute__((ext_vector_type(8)));
typedef int   v4i __attribute__((vector_size(4 * sizeof(int))));

// Workspace layout (in floats), all 16-float aligned
#define OFF_EMBN   0            // 64*64      normalized circuit_emb[:64]
#define OFF_BIAS   4096         // 64         excitability * exc
#define OFF_NEMBN  4160         // 64*64*64   normalized neuron_emb
#define OFF_H2     266304       // 4096*64    h2 = x @ neuron_proj_W.T
#define OFF_LOGITS 528448       // 4096*64    biased circuit logits

// --- CDNA5 async global->LDS support (ASYNCcnt-tracked DMA path) -----------
#if defined(__has_builtin)
#if __has_builtin(__builtin_amdgcn_global_load_async_to_lds_b128)
#define HAVE_ASYNC_LDS 1
#endif
#endif

__device__ __forceinline__ void cr_wait_asynccnt0() {
#if defined(__has_builtin) && __has_builtin(__builtin_amdgcn_s_wait_asynccnt)
  __builtin_amdgcn_s_wait_asynccnt(0);
#else
  asm volatile("s_wait_asynccnt 0x0" ::: "memory");
#endif
}

// Copy 16 bytes global -> LDS. Async variant uses GLOBAL_LOAD_ASYNC_TO_LDS_B128.
// Builtin signature (from hipcc diagnostic): (v4i __device__*, v4i __shared__*,
// imm offset, imm cpol); generic pointers implicitly convert in HIP mode.
__device__ __forceinline__ void cr_copy16_g2l(float* dstLds, const float* srcGlobal) {
#ifdef HAVE_ASYNC_LDS
  __builtin_amdgcn_global_load_async_to_lds_b128(
      (v4i*)srcGlobal, (v4i*)dstLds, 0, 0);
#else
  *reinterpret_cast<float4*>(dstLds) =
      *reinterpret_cast<const float4*>(srcGlobal);
#endif
}

// ---------------------------------------------------------------------------
// Kernel 1: normalize embedding tables + excitability bias
// grid = 65 blocks x 64 threads.  blocks 0..63: one circuit's 64 neurons;
// block 64: circuit_emb rows + bias.
// ---------------------------------------------------------------------------
__global__ __launch_bounds__(64)
void cr_prep_kernel(const float* __restrict__ circuit_emb,
                    const float* __restrict__ neuron_emb,
                    const float* __restrict__ usage,
                    const float* __restrict__ exc,
                    float* __restrict__ ws) {
  const int t = threadIdx.x;  // 0..63
  if (blockIdx.x < NC_R) {
    const int c = blockIdx.x;
    const float* src = neuron_emb + ((size_t)c * N_NEUR + t) * D_SPACE;
    float* dst = ws + OFF_NEMBN + ((size_t)c * N_NEUR + t) * D_SPACE;
    float ss = 0.f;
    #pragma unroll 8
    for (int d = 0; d < D_SPACE; ++d) { float v = src[d]; ss += v * v; }
    const float s = 1.f / fmaxf(sqrtf(ss), 1e-12f);
    #pragma unroll 8
    for (int d = 0; d < D_SPACE; ++d) dst[d] = src[d] * s;
  } else {
    const float* src = circuit_emb + (size_t)t * D_SPACE;   // first 64 rows only
    float* dst = ws + OFF_EMBN + (size_t)t * D_SPACE;
    float ss = 0.f;
    #pragma unroll 8
    for (int d = 0; d < D_SPACE; ++d) { float v = src[d]; ss += v * v; }
    const float s = 1.f / fmaxf(sqrtf(ss), 1e-12f);
    #pragma unroll 8
    for (int d = 0; d < D_SPACE; ++d) dst[d] = src[d] * s;
    float ex = 1.f - usage[t] * TAU_INV;
    ex = fminf(fmaxf(ex, 0.f), 1.f);
    ws[OFF_BIAS + t] = ex * exc[t];
  }
}

// ---------------------------------------------------------------------------
// Kernel 2: fused projection GEMM + logits GEMM (V_WMMA_F32_16X16X4_F32)
// grid = 256 blocks (16 rows each) x 256 threads (8 waves).
// Wave w owns output columns [w*16, w*16+16) of the fused 128-wide result
// (cols 0..63 = h, cols 64..127 = h2).  x tile double-buffered through LDS
// via the async global->LDS DMA path; WMMAs overlap the next chunk's fill.
// ---------------------------------------------------------------------------
#define KC        256
#define NCHUNK    (D_MODEL / KC)
#define XS_STRIDE 260   // 16x256 chunk, +4 pad -> conflict-free ds_load_b64
#define HS_STRIDE 68    // 16x64 h tile, +4 pad

__global__ __launch_bounds__(256)
void cr_proj_gemm_kernel(const float* __restrict__ x,
                         const float* __restrict__ projW,
                         const float* __restrict__ nprojW,
                         float* __restrict__ ws) {
  __shared__ float xs[2][16 * XS_STRIDE];
  __shared__ float hs[16 * HS_STRIDE];

  const float* embn = ws + OFF_EMBN;
  const float* bias = ws + OFF_BIAS;
  float* h2g  = ws + OFF_H2;
  float* logg = ws + OFF_LOGITS;

  const int tid  = threadIdx.x;
  const int wv   = tid >> 5;
  const int lane = tid & 31;
  const int hf   = lane >> 4;   // which K-pair half (ISA A/B layout)
  const int l16  = lane & 15;
  const int mbase = blockIdx.x * 16;

  const float* Wsel = (wv < 4) ? projW : nprojW;
  const float* wrow = Wsel + (size_t)((wv & 3) * 16 + l16) * D_MODEL;

  // This thread's fixed slice of each 16x256 staging chunk (4 x 16B).
  int srow[4], scol[4];
  #pragma unroll
  for (int i = 0; i < 4; ++i) {
    const int f4 = tid + i * 256;         // 1024 float4s per chunk
    srow[i] = f4 >> 6;
    scol[i] = (f4 & 63) << 2;
  }

  // Prologue: fill buffer 0 with chunk 0.
  #pragma unroll
  for (int i = 0; i < 4; ++i)
    cr_copy16_g2l(&xs[0][srow[i] * XS_STRIDE + scol[i]],
                  x + (size_t)(mbase + srow[i]) * D_MODEL + scol[i]);
#ifdef HAVE_ASYNC_LDS
  cr_wait_asynccnt0();
#endif
  __syncthreads();

  v8f acc = {};
  int buf = 0;

  for (int c = 0; c < NCHUNK; ++c) {
    const int kc = c * KC;
    // Kick off async fill of the next chunk into the other buffer.
    if (c + 1 < NCHUNK) {
      #pragma unroll
      for (int i = 0; i < 4; ++i)
        cr_copy16_g2l(&xs[buf ^ 1][srow[i] * XS_STRIDE + scol[i]],
                      x + (size_t)(mbase + srow[i]) * D_MODEL + kc + KC + scol[i]);
    }

    // Consume current chunk: 64 x V_WMMA_F32_16X16X4_F32.
    const float2* xrow2 =
        reinterpret_cast<const float2*>(&xs[buf][l16 * XS_STRIDE + 2 * hf]);
    const float2* wrow2 = reinterpret_cast<const float2*>(wrow + kc + 2 * hf);
    #pragma unroll 8
    for (int k = 0; k < KC; k += 4) {
      const float2 af = xrow2[k >> 1];   // A: M=l16, K = k + 2*hf {,+1}
      const float2 bf = wrow2[k >> 1];   // B: N=l16, K = k + 2*hf {,+1}
      v2f a; a.x = af.x; a.y = af.y;
      v2f b; b.x = bf.x; b.y = bf.y;
      acc = __builtin_amdgcn_wmma_f32_16x16x4_f32(
          false, a, false, b, (short)0, acc, false, false);
    }

#ifdef HAVE_ASYNC_LDS
    cr_wait_asynccnt0();   // my async fills of buf^1 are complete
#endif
    __syncthreads();       // everyone's fills visible; safe to swap
    buf ^= 1;
  }

  // C/D layout: VGPR r holds (M = r + 8*hf, N = l16)
  if (wv >= 4) {
    const int col = (wv - 4) * 16 + l16;
    #pragma unroll
    for (int r = 0; r < 8; ++r)
      h2g[(size_t)(mbase + r + 8 * hf) * D_SPACE + col] = acc[r];
  } else {
    const int col = wv * 16 + l16;
    #pragma unroll
    for (int r = 0; r < 8; ++r)
      hs[(r + 8 * hf) * HS_STRIDE + col] = acc[r];
  }
  __syncthreads();

  // Second WMMA pass: logits = h @ emb_norm.T + bias  (16x64, K=64)
  if (wv < 4) {
    v8f lacc = {};
    const int cc = wv * 16 + l16;  // circuit column
    const float2* hrow2 =
        reinterpret_cast<const float2*>(&hs[l16 * HS_STRIDE + 2 * hf]);
    const float2* erow2 =
        reinterpret_cast<const float2*>(embn + (size_t)cc * D_SPACE + 2 * hf);
    #pragma unroll
    for (int k = 0; k < D_SPACE; k += 4) {
      const float2 af = hrow2[k >> 1];
      const float2 bf = erow2[k >> 1];
      v2f a; a.x = af.x; a.y = af.y;
      v2f b; b.x = bf.x; b.y = bf.y;
      lacc = __builtin_amdgcn_wmma_f32_16x16x4_f32(
          false, a, false, b, (short)0, lacc, false, false);
    }
    const float bv = bias[cc];
    #pragma unroll
    for (int r = 0; r < 8; ++r)
      logg[(size_t)(mbase + r + 8 * hf) * NC_R + cc] = lacc[r] + bv;
  }
}

// ---------------------------------------------------------------------------
// Kernel 3: per-row top-8 + gathered neuron logits + softmax.
// One wave32 per batch row; 8 waves/block; grid = 512.
// ---------------------------------------------------------------------------
__global__ __launch_bounds__(256)
void cr_route_kernel(float* __restrict__ out, const float* __restrict__ ws) {
  __shared__ float h2s[8][64];

  const float* nembn = ws + OFF_NEMBN;
  const float* h2g   = ws + OFF_H2;
  const float* logg  = ws + OFF_LOGITS;

  const int tid  = threadIdx.x;
  const int wv   = tid >> 5;
  const int lane = tid & 31;
  const int b    = blockIdx.x * 8 + wv;

  // Stage this row's h2 into LDS (wave-private; same-wave LDS ops are in-order)
  h2s[wv][lane]      = h2g[(size_t)b * D_SPACE + lane];
  h2s[wv][lane + 32] = h2g[(size_t)b * D_SPACE + lane + 32];

  float v0 = logg[(size_t)b * NC_R + lane];        // circuit idx = lane
  float v1 = logg[(size_t)b * NC_R + lane + 32];   // circuit idx = lane+32

  const float4* h4 = reinterpret_cast<const float4*>(&h2s[wv][0]);
  float* outb = out + (size_t)b * TOPK * N_NEUR;

  for (int kk = 0; kk < TOPK; ++kk) {
    // --- wave-wide argmax over 64 logits (tie -> lowest index, like top_k) ---
    float bv; int bi;
    if (v1 > v0) { bv = v1; bi = lane + 32; } else { bv = v0; bi = lane; }
    #pragma unroll
    for (int off = 16; off > 0; off >>= 1) {
      const float ov = __shfl_xor(bv, off);
      const int   oi = __shfl_xor(bi, off);
      if (ov > bv || (ov == bv && oi < bi)) { bv = ov; bi = oi; }
    }
    const int c = bi;                     // uniform across wave
    if (c == lane)      v0 = -3.4e38f;    // knock out selected circuit
    if (c == lane + 32) v1 = -3.4e38f;

    // --- inner logits: h2 . nembn[c, n, :] for n = lane, lane+32 ---
    const float4* w0 = reinterpret_cast<const float4*>(
        nembn + ((size_t)c * N_NEUR + lane) * D_SPACE);
    const float4* w1 = reinterpret_cast<const float4*>(
        nembn + ((size_t)c * N_NEUR + lane + 32) * D_SPACE);
    float a0 = 0.f, a1 = 0.f;
    #pragma unroll
    for (int d = 0; d < D_SPACE / 4; ++d) {
      const float4 h  = h4[d];          // LDS broadcast (same addr all lanes)
      const float4 e0 = w0[d];
      const float4 e1 = w1[d];
      a0 += h.x * e0.x + h.y * e0.y + h.z * e0.z + h.w * e0.w;
      a1 += h.x * e1.x + h.y * e1.y + h.z * e1.z + h.w * e1.w;
    }

    // --- softmax over 64 neurons ---
    float m = fmaxf(a0, a1);
    #pragma unroll
    for (int off = 16; off > 0; off >>= 1) m = fmaxf(m, __shfl_xor(m, off));
    const float e0v = __expf(a0 - m);
    const float e1v = __expf(a1 - m);
    float s = e0v + e1v;
    #pragma unroll
    for (int off = 16; off > 0; off >>= 1) s += __shfl_xor(s, off);
    const float inv = 1.f / s;
    outb[kk * N_NEUR + lane]      = e0v * inv;
    outb[kk * N_NEUR + lane + 32] = e1v * inv;
  }
}

// ---------------------------------------------------------------------------
extern "C" void kernel_launch(void* const* d_in, const int* in_sizes, int n_in,
                              void* d_out, int out_size, void* d_ws, size_t ws_size,
                              hipStream_t stream) {
  const float* x           = (const float*)d_in[0];  // [4096,1024]
  const float* projW       = (const float*)d_in[1];  // [64,1024]
  const float* circuit_emb = (const float*)d_in[2];  // [320,64]
  const float* nprojW      = (const float*)d_in[3];  // [64,1024]
  const float* neuron_emb  = (const float*)d_in[4];  // [64,64,64]
  const float* usage       = (const float*)d_in[5];  // [64]
  const float* exc         = (const float*)d_in[6];  // [64]
  (void)in_sizes; (void)n_in; (void)out_size; (void)ws_size;  // k fixed at 8

  float* ws  = (float*)d_ws;
  float* out = (float*)d_out;

  cr_prep_kernel<<<NC_R + 1, 64, 0, stream>>>(circuit_emb, neuron_emb, usage, exc, ws);
  cr_proj_gemm_kernel<<<B_ROWS / 16, 256, 0, stream>>>(x, projW, nprojW, ws);
  cr_route_kernel<<<B_ROWS / 8, 256, 0, stream>>>(out, ws);
}